// Wavelet_Mahalanobis_mask_84817014162118
// MI455X (gfx1250) — compile-verified
//
#include <hip/hip_runtime.h>
#include <hip/hip_bf16.h>
#include <math.h>

typedef float v2f __attribute__((ext_vector_type(2)));
typedef float v8f __attribute__((ext_vector_type(8)));

#define B_DIM 16
#define N_DIM 1024
#define L_DIM 96
#define NROWS (B_DIM * N_DIM)      // 16384
#define INV_SQRT2 0.70710678118654752440f
#define LN2 0.69314718055994530942f

// ---------------------------------------------------------------------------
// K1: Haar DWT (3 levels) + mean/var normalize + abs  ->  Xw [16384 x 96]
// One row per thread; tiny (6 MB) kernel.
// ---------------------------------------------------------------------------
__global__ void haar_norm_kernel(const float* __restrict__ X, float* __restrict__ Xw) {
    int row = blockIdx.x * blockDim.x + threadIdx.x;
    if (row >= NROWS) return;
    const float* x = X + (size_t)row * L_DIM;

    float W[96];
    float s1[48], s2[24];
    #pragma unroll
    for (int i = 0; i < 48; ++i) {
        float e = x[2 * i], o = x[2 * i + 1];
        W[i]  = (e - o) * INV_SQRT2;       // detail level 1 -> W[0..47]
        s1[i] = (e + o) * INV_SQRT2;
    }
    #pragma unroll
    for (int i = 0; i < 24; ++i) {
        float e = s1[2 * i], o = s1[2 * i + 1];
        W[48 + i] = (e - o) * INV_SQRT2;   // detail level 2 -> W[48..71]
        s2[i]     = (e + o) * INV_SQRT2;
    }
    #pragma unroll
    for (int i = 0; i < 12; ++i) {
        float e = s2[2 * i], o = s2[2 * i + 1];
        W[72 + i] = (e - o) * INV_SQRT2;   // detail level 3 -> W[72..83]
        W[84 + i] = (e + o) * INV_SQRT2;   // approx         -> W[84..95]
    }

    float sum = 0.f;
    #pragma unroll
    for (int i = 0; i < 96; ++i) sum += W[i];
    float mu = sum * (1.0f / 96.0f);
    float vs = 0.f;
    #pragma unroll
    for (int i = 0; i < 96; ++i) { float d = W[i] - mu; vs += d * d; }
    float var = vs * (1.0f / 96.0f);
    float inv = rsqrtf(var + 1e-05f);

    float* o = Xw + (size_t)row * L_DIM;
    #pragma unroll
    for (int i = 0; i < 96; ++i) o[i] = fabsf((W[i] - mu) * inv);
}

// ---------------------------------------------------------------------------
// K2: Y = Xw * A^T  via V_WMMA_F32_16X16X4_F32.
// Y[r,k] = sum_d Xw[r,d] * A[k,d].  GEMM A-operand = Xw rows, B-operand[d,k] = A[k,d].
// One 16x16 output tile per wave; 24 K-steps of 4.
// A-frag layout (ISA 7.12.2, 32-bit A 16x4): lane(lk,half): VGPR j holds K = j + 2*half, M = lk.
// C/D layout: VGPR j holds row (j + 8*half), col = lk.
// ---------------------------------------------------------------------------
__global__ void ygemm_kernel(const float* __restrict__ Xw, const float* __restrict__ A,
                             float* __restrict__ Y) {
    int wave = threadIdx.x >> 5;
    int lane = threadIdx.x & 31;
    int lk   = lane & 15;
    int half = lane >> 4;
    int tile = blockIdx.x * 8 + wave;           // 6144 tiles = 1024 row-tiles * 6 col-tiles
    if (tile >= (NROWS / 16) * (L_DIM / 16)) return;
    int rt = tile / (L_DIM / 16);
    int ct = tile % (L_DIM / 16);

    v8f acc = {};
    const float* arow = Xw + ((size_t)(rt * 16 + lk)) * L_DIM;     // GEMM-A: Xw row lk
    const float* brow = A  + ((size_t)(ct * 16 + lk)) * L_DIM;     // GEMM-B col lk = A row (k)
    #pragma unroll
    for (int kb = 0; kb < 24; ++kb) {
        int d0 = kb * 4 + 2 * half;
        v2f a = *(const v2f*)(arow + d0);
        v2f b = *(const v2f*)(brow + d0);
        acc = __builtin_amdgcn_wmma_f32_16x16x4_f32(false, a, false, b,
                                                    (short)0, acc, false, false);
    }
    #pragma unroll
    for (int j = 0; j < 8; ++j) {
        int m = rt * 16 + j + 8 * half;
        Y[(size_t)m * L_DIM + ct * 16 + lk] = acc[j];
    }
}

// ---------------------------------------------------------------------------
// K2b: sq[r] = sum_k Y[r,k]^2
// ---------------------------------------------------------------------------
__global__ void sq_kernel(const float* __restrict__ Y, float* __restrict__ sq) {
    int r = blockIdx.x * blockDim.x + threadIdx.x;
    if (r >= NROWS) return;
    const float* y = Y + (size_t)r * L_DIM;
    float s = 0.f;
    #pragma unroll
    for (int k = 0; k < 96; ++k) s += y[k] * y[k];
    sq[r] = s;
}

// ---------------------------------------------------------------------------
// K3: per (batch, 16-row strip): G strip = Yrows(16x96) * Ybatch^T (96x1024) via
// f32 WMMA, fused with dist -> e = 1/(dist+1e-5) (diag=0), rowmax, p, and the
// Gumbel argmax threshold. e strip (16x1024) lives entirely in registers
// (8 waves x 8 tiles x v8f). Output is hard0 in {0,1}.
// ---------------------------------------------------------------------------
__global__ void __launch_bounds__(256)
dist_mask_kernel(const float* __restrict__ Y, const float* __restrict__ sq,
                 const float2* __restrict__ gumbel, float* __restrict__ out) {
    __shared__ float ldsY[16 * 96];       // the 16-row Y strip
    __shared__ float ldsPart[8 * 16];     // per-wave row maxima
    __shared__ float ldsRowMax[16];

    int batch = blockIdx.y;               // 0..15
    int rtile = blockIdx.x;               // 0..63 (16-row strips)
    int wave  = threadIdx.x >> 5;
    int lane  = threadIdx.x & 31;
    int lk    = lane & 15;
    int half  = lane >> 4;

    size_t batchRow = (size_t)batch * N_DIM;
    size_t stripRow = batchRow + rtile * 16;

    // cooperative load of the 16x96 Y strip (flat, coalesced)
    {
        const float* src = Y + stripRow * L_DIM;
        for (int i = threadIdx.x; i < 16 * 96; i += 256) ldsY[i] = src[i];
    }
    __syncthreads();

    // per-lane row sq (row depends on j+8*half only; broadcast loads)
    float sqr[8];
    #pragma unroll
    for (int j = 0; j < 8; ++j) sqr[j] = sq[stripRow + j + 8 * half];

    v8f accT[8];
    float pmax[8];
    #pragma unroll
    for (int j = 0; j < 8; ++j) pmax[j] = 0.f;

    // Phase A: 8 column tiles per wave; tile t covers cols (wave + 8*t)*16
    #pragma unroll
    for (int t = 0; t < 8; ++t) {
        int n0 = (wave + 8 * t) * 16;
        const float* arow = ldsY + lk * 96;
        const float* brow = Y + (batchRow + n0 + lk) * L_DIM;   // GEMM-B col = Y row (n0+lk)
        v8f acc = {};
        #pragma unroll
        for (int kb = 0; kb < 24; ++kb) {
            int d0 = kb * 4 + 2 * half;
            v2f a = *(const v2f*)(arow + d0);
            v2f b = *(const v2f*)(brow + d0);
            acc = __builtin_amdgcn_wmma_f32_16x16x4_f32(false, a, false, b,
                                                        (short)0, acc, false, false);
        }
        float sqc = sq[batchRow + n0 + lk];
        #pragma unroll
        for (int j = 0; j < 8; ++j) {
            int rrow = rtile * 16 + j + 8 * half;   // row index within batch
            int col  = n0 + lk;
            float dist = sqr[j] + sqc - 2.0f * acc[j];
            dist = fmaxf(dist, 0.0f);
            float e = 1.0f / (dist + 1e-05f);
            if (col == rrow) e = 0.0f;              // off-diagonal mask
            acc[j] = e;
            pmax[j] = fmaxf(pmax[j], e);
        }
        accT[t] = acc;
    }

    // cross-lane row max within each 16-lane half (rows j+8*half)
    #pragma unroll
    for (int j = 0; j < 8; ++j) {
        float m = pmax[j];
        m = fmaxf(m, __shfl_xor(m, 1, 32));
        m = fmaxf(m, __shfl_xor(m, 2, 32));
        m = fmaxf(m, __shfl_xor(m, 4, 32));
        m = fmaxf(m, __shfl_xor(m, 8, 32));
        pmax[j] = m;
        if (lk == 0) ldsPart[wave * 16 + j + 8 * half] = m;
    }
    __syncthreads();
    if (threadIdx.x < 16) {
        float m = 0.f;
        #pragma unroll
        for (int w = 0; w < 8; ++w) m = fmaxf(m, ldsPart[w * 16 + threadIdx.x]);
        ldsRowMax[threadIdx.x] = m;
    }
    __syncthreads();

    float rminv[8];
    #pragma unroll
    for (int j = 0; j < 8; ++j)
        rminv[j] = 1.0f / (ldsRowMax[j + 8 * half] + 1e-05f);

    // Phase B: p, logits, gumbel threshold, write 0/1 output
    #pragma unroll
    for (int t = 0; t < 8; ++t) {
        int n0 = (wave + 8 * t) * 16;
        #pragma unroll
        for (int j = 0; j < 8; ++j) {
            int rrow = rtile * 16 + j + 8 * half;
            int col  = n0 + lk;
            float p = (col == rrow) ? 0.99f : 0.99f * accT[t][j] * rminv[j];
            // argmax==0  <=>  log(p+1e-10)+g0 >= log(1-p+1e-10)+g1
            float ldiff = LN2 * (__log2f(p + 1e-10f) - __log2f(1.0f - p + 1e-10f));
            float2 g = gumbel[(batchRow + rrow) * N_DIM + col];
            out[(batchRow + rrow) * N_DIM + col] = (ldiff >= (g.y - g.x)) ? 1.0f : 0.0f;
        }
    }
}

// ---------------------------------------------------------------------------
extern "C" void kernel_launch(void* const* d_in, const int* in_sizes, int n_in,
                              void* d_out, int out_size, void* d_ws, size_t ws_size,
                              hipStream_t stream) {
    const float* X = (const float*)d_in[0];       // (16,1024,96)
    const float* A = (const float*)d_in[1];       // (96,96)
    const float2* gumbel = (const float2*)d_in[2];// (16,1024,1024,2)
    float* out = (float*)d_out;                   // (16,1,1024,1024)

    // workspace layout
    float* Xw = (float*)d_ws;                     // 16384*96
    float* Y  = Xw + (size_t)NROWS * L_DIM;       // 16384*96
    float* sq = Y  + (size_t)NROWS * L_DIM;       // 16384

    haar_norm_kernel<<<NROWS / 256, 256, 0, stream>>>(X, Xw);

    // 6144 tiles, 8 waves (tiles) per 256-thread block
    ygemm_kernel<<<(NROWS / 16) * (L_DIM / 16) / 8, 256, 0, stream>>>(Xw, A, Y);

    sq_kernel<<<NROWS / 256, 256, 0, stream>>>(Y, sq);

    dim3 grid(N_DIM / 16, B_DIM);
    dist_mask_kernel<<<grid, 256, 0, stream>>>(Y, sq, gumbel, out);
}